// CustomLoss_Z_B_25031069401264
// MI455X (gfx1250) — compile-verified
//
#include <hip/hip_runtime.h>
#include <math.h>

// ---------------------------------------------------------------------------
// Geometry constants (dx = dy = dz = 1 after normalization in the reference)
// ---------------------------------------------------------------------------
#define NG 96
#define NC 95
static constexpr int V       = NG * NG * NG;     // 884736 voxels per channel
static constexpr int NCELL   = NC * NC * NC;     // 857375 div cells
static constexpr int NJ      = NG * NC * NC;     // 866400 elements in each j field
#define DIV_BLOCKS 3392                          // ceil(857375/256)=3350, padded to 53*64
#define MED_BLOCKS 1728                          // 12^3 tiles of 8^3 (= 27*64)

typedef __attribute__((ext_vector_type(2))) float v2f;
typedef __attribute__((ext_vector_type(8))) float v8f;

#if defined(__has_builtin)
#if __has_builtin(__builtin_amdgcn_wmma_f32_16x16x4_f32)
#define HAVE_WMMA_F32 1
#endif
#endif

__device__ __forceinline__ int refl(int i, int n) {
    i = (i < 0) ? -i : i;                 // 'reflect' mode: edge not repeated
    return (i >= n) ? (2 * n - 2 - i) : i;
}

#define CSWAP(x, y) do { float _lo = fminf((x), (y)); float _hi = fmaxf((x), (y)); (x) = _lo; (y) = _hi; } while (0)

// ---------------------------------------------------------------------------
// Forgetful selection, fully unrolled via template recursion so every buffer
// index is a compile-time constant (keeps b[64] entirely in VGPRs, no guards).
// One "drop pass" over the first S entries: forward sweep pushes max to
// b[S-1], backward sweep pushes min to b[0]; both get discarded.
// ---------------------------------------------------------------------------
template <int S>
__device__ __forceinline__ void drop_pass(float (&b)[64]) {
#pragma unroll
    for (int i = 0; i < S - 1; ++i) CSWAP(b[i], b[i + 1]);
#pragma unroll
    for (int i = S - 2; i >= 1; --i) CSWAP(b[i - 1], b[i]);
}

template <int E>
struct Stream {
    static __device__ __forceinline__ void run(float (&b)[64], const float* tile, int base) {
        drop_pass<64 - E>(b);                 // current logical size = 64 - E
        constexpr int wi  = 64 + E;           // stream window element 64..124
        constexpr int dz  = wi / 25;
        constexpr int r   = wi % 25;
        constexpr int dy  = r / 5;
        constexpr int dxo = r % 5;
        b[0] = tile[base + dz * 144 + dy * 12 + dxo];   // overwrite dropped min
        Stream<E + 1>::run(b, tile, base);
    }
};
template <>
struct Stream<61> {
    static __device__ __forceinline__ void run(float (&)[64], const float*, int) {}
};

// ---------------------------------------------------------------------------
// Kernel 1: mask & masked target fields  bxm = bx_t*mask, bym = by_t*mask
// ---------------------------------------------------------------------------
__global__ void prep_kernel(const float* __restrict__ pb, const float* __restrict__ tg,
                            float* __restrict__ bxm, float* __restrict__ bym) {
    int i = blockIdx.x * blockDim.x + threadIdx.x;
    if (i >= V) return;
    float bxp = pb[i], byp = pb[V + i];
    float bxt = tg[i], byt = tg[V + i];
    float m = (bxp * bxt + byp * byt > 0.f) ? 1.f : -1.f;
    bxm[i] = bxt * m;
    bym[i] = byt * m;
}

// ---------------------------------------------------------------------------
// Kernel 2: current density jx (96,95,95), jy (95,96,95), jz (95,95,96)
// ---------------------------------------------------------------------------
__global__ void j_kernel(const float* __restrict__ bxm, const float* __restrict__ bym,
                         const float* __restrict__ bzp,
                         float* __restrict__ jx, float* __restrict__ jy, float* __restrict__ jz) {
    int i = blockIdx.x * blockDim.x + threadIdx.x;
    if (i >= V) return;
    int w = i % NG, h = (i / NG) % NG, d = i / (NG * NG);
#define AT(p, dd, hh, ww) (p)[((dd) * NG + (hh)) * NG + (ww)]
    if (h < NC && w < NC) {
        float dyBz0 = AT(bzp, d, h, w)     - AT(bzp, d, h + 1, w);
        float dyBz1 = AT(bzp, d, h, w + 1) - AT(bzp, d, h + 1, w + 1);
        float dzBy0 = AT(bym, d, h, w)     - AT(bym, d, h, w + 1);
        float dzBy1 = AT(bym, d, h + 1, w) - AT(bym, d, h + 1, w + 1);
        jx[(d * NC + h) * NC + w] = 0.5f * (dyBz0 + dyBz1) - 0.5f * (dzBy0 + dzBy1);
    }
    if (d < NC && w < NC) {
        float dzBx0 = AT(bxm, d, h, w)     - AT(bxm, d, h, w + 1);
        float dzBx1 = AT(bxm, d + 1, h, w) - AT(bxm, d + 1, h, w + 1);
        float dxBz0 = AT(bzp, d, h, w)     - AT(bzp, d + 1, h, w);
        float dxBz1 = AT(bzp, d, h, w + 1) - AT(bzp, d + 1, h, w + 1);
        jy[(d * NG + h) * NC + w] = 0.5f * (dzBx0 + dzBx1) - 0.5f * (dxBz0 + dxBz1);
    }
    if (d < NC && h < NC) {
        float dxBy0 = AT(bym, d, h, w)     - AT(bym, d + 1, h, w);
        float dxBy1 = AT(bym, d, h + 1, w) - AT(bym, d + 1, h + 1, w);
        float dyBx0 = AT(bxm, d, h, w)     - AT(bxm, d, h + 1, w);
        float dyBx1 = AT(bxm, d + 1, h, w) - AT(bxm, d + 1, h + 1, w);
        jz[(d * NC + h) * NG + w] = 0.5f * (dxBy0 + dxBy1) - 0.5f * (dyBx0 + dyBx1);
    }
#undef AT
}

// ---------------------------------------------------------------------------
// flux^4 / ave_b for one 2x2x2 cell; corner index c = i*4 + j*2 + l
// ---------------------------------------------------------------------------
__device__ __forceinline__ float flx1_of(const float* bx, const float* by,
                                         const float* bz, const float* z) {
    const int c000 = 0, c001 = 1, c010 = 2, c011 = 3, c100 = 4, c101 = 5, c110 = 6, c111 = 7;
    const float third = 1.f / 3.f, sixth = 1.f / 6.f;
    float flux =
        0.25f * (bx[c100] + bx[c110] + bx[c101] + bx[c111]) * 0.5f * (fabsf(z[c101] - z[c100]) + fabsf(z[c111] - z[c110]))
      - 0.25f * (bx[c000] + bx[c010] + bx[c001] + bx[c011]) * 0.5f * (fabsf(z[c001] - z[c000]) + fabsf(z[c011] - z[c010]))
      + 0.25f * (by[c010] + by[c110] + by[c011] + by[c111]) * 0.5f * (fabsf(z[c011] - z[c010]) + fabsf(z[c111] - z[c110]))
      - 0.25f * (by[c000] + by[c100] + by[c001] + by[c101]) * 0.5f * (fabsf(z[c001] - z[c000]) + fabsf(z[c101] - z[c100]))
      + 0.5f * ((bz[c001] + bz[c101] + bz[c111]) * third + (bz[c001] + bz[c111] + bz[c011]) * third)
      - 0.5f * ((bz[c000] + bz[c100] + bz[c110]) * third + (bz[c000] + bz[c110] + bz[c010]) * third)
      + (bx[c001] + bx[c101] + bx[c111]) * (z[c001] - z[c101]) * sixth
      + (bx[c001] + bx[c011] + bx[c111]) * (z[c011] - z[c111]) * sixth
      + (by[c001] + by[c101] + by[c111]) * (z[c101] - z[c111]) * sixth
      + (by[c001] + by[c011] + by[c111]) * (z[c001] - z[c011]) * sixth
      - ( (bx[c000] + bx[c100] + bx[c110]) * (z[c000] - z[c100]) * sixth
        + (bx[c000] + bx[c010] + bx[c110]) * (z[c010] - z[c110]) * sixth
        + (by[c000] + by[c100] + by[c110]) * (z[c100] - z[c110]) * sixth
        + (by[c000] + by[c010] + by[c110]) * (z[c000] - z[c010]) * sixth );
    float sx = 0.f, sy = 0.f, sz = 0.f;
#pragma unroll
    for (int c = 0; c < 8; ++c) { sx += bx[c]; sy += by[c]; sz += bz[c]; }
    sx *= 0.125f; sy *= 0.125f; sz *= 0.125f;
    float ave = sx * sx + sy * sy + sz * sz + 1e-8f;
    float r = flux * flux;
    return (r * r) / ave;
}

// ---------------------------------------------------------------------------
// Kernel 3: cal_div_c_old for both variants; per-block partial Σx, Σx²
//           partials: [S1_p | S2_p | S1_t | S2_t], each DIV_BLOCKS entries.
//           Grid is padded to DIV_BLOCKS so pad blocks write exact zeros and
//           the final reduction needs no tail handling.
// ---------------------------------------------------------------------------
__global__ __launch_bounds__(256) void div_kernel(const float* __restrict__ pb,
                                                  const float* __restrict__ pz,
                                                  const float* __restrict__ tg,
                                                  float* __restrict__ part) {
    __shared__ float red[256];
    int tid = threadIdx.x;
    int idx = blockIdx.x * 256 + tid;
    float acc[4] = {0.f, 0.f, 0.f, 0.f};
    if (idx < NCELL) {
        int w = idx % NC, h = (idx / NC) % NC, d = idx / (NC * NC);
        const float* bxp = pb;
        const float* byp = pb + V;
        const float* bxt = tg;
        const float* byt = tg + V;
        const float* bzt = tg + 2 * V;
        float BXP[8], BYP[8], BXM[8], BYM[8], BZ[8], ZC[8];
#pragma unroll
        for (int c = 0; c < 8; ++c) {
            int ci = (c >> 2) & 1, cj = (c >> 1) & 1, cl = c & 1;
            int o = ((d + ci) * NG + (h + cj)) * NG + (w + cl);
            float xp = bxp[o], yp = byp[o], xt = bxt[o], yt = byt[o];
            float m = (xp * xt + yp * yt > 0.f) ? 1.f : -1.f;
            BXP[c] = xp; BYP[c] = yp;
            BXM[c] = xt * m; BYM[c] = yt * m;
            BZ[c] = bzt[o];
            ZC[c] = pz[o];                      // z = pred_z / dx, dx == 1
        }
        float fp = flx1_of(BXP, BYP, BZ, ZC);
        float ft = flx1_of(BXM, BYM, BZ, ZC);
        acc[0] = fp; acc[1] = fp * fp;
        acc[2] = ft; acc[3] = ft * ft;
    }
#pragma unroll
    for (int k = 0; k < 4; ++k) {
        red[tid] = acc[k];
        __syncthreads();
        for (int s = 128; s > 0; s >>= 1) {
            if (tid < s) red[tid] += red[tid + s];
            __syncthreads();
        }
        if (tid == 0) part[k * DIV_BLOCKS + blockIdx.x] = red[0];
        __syncthreads();
    }
}

// ---------------------------------------------------------------------------
// Kernel 4: 5x5x5 median filter (reflect) + per-block Σ(med - x)²
// 8³ outputs per block (16 wave32s), 12³ LDS halo tile, straight-line
// forgetful selection in 64 VGPRs (template-forced full unroll).
// ---------------------------------------------------------------------------
__global__ __launch_bounds__(512) void med_kernel(const float* __restrict__ src,
                                                  int D, int H, int W,
                                                  float* __restrict__ part) {
    __shared__ float tile[12 * 12 * 12];
    __shared__ float red[512];
    int t = threadIdx.x;
    int bx0 = blockIdx.x * 8, by0 = blockIdx.y * 8, bz0 = blockIdx.z * 8;
    for (int i = t; i < 1728; i += 512) {
        int lx = i % 12, ly = (i / 12) % 12, lz = i / 144;
        int gz = refl(bz0 + lz - 2, D);
        int gy = refl(by0 + ly - 2, H);
        int gx = refl(bx0 + lx - 2, W);
        tile[i] = src[(gz * H + gy) * W + gx];
    }
    __syncthreads();

    int lx = t & 7, ly = (t >> 3) & 7, lz = t >> 6;
    int gx = bx0 + lx, gy = by0 + ly, gz = bz0 + lz;
    float val = 0.f;
    if (gx < W && gy < H && gz < D) {
        int base = lz * 144 + ly * 12 + lx;
        float b[64];
#pragma unroll
        for (int i = 0; i < 64; ++i) {
            int dz = i / 25, r = i - dz * 25, dy = r / 5, dxo = r - dy * 5;
            b[i] = tile[base + dz * 144 + dy * 12 + dxo];
        }
        Stream<0>::run(b, tile, base);        // stream elements 64..124
        float med = fmaxf(fminf(b[0], b[1]), fminf(fmaxf(b[0], b[1]), b[2]));
        float x = tile[base + 2 * 144 + 2 * 12 + 2];
        float dv = med - x;
        val = dv * dv;
    }
    red[t] = val;
    __syncthreads();
    for (int s = 256; s > 0; s >>= 1) {
        if (t < s) red[t] += red[t + s];
        __syncthreads();
    }
    if (t == 0) {
        int bid = (blockIdx.z * gridDim.y + blockIdx.y) * gridDim.x + blockIdx.x;
        part[bid] = red[0];
    }
}

// ---------------------------------------------------------------------------
// Kernel 5: final deterministic reduction of 11 partial arrays.
// One wave32 (EXEC all ones as WMMA requires). All array lengths are
// multiples of 64: each lane does one unconditional b64 load (2 consecutive
// floats) per V_WMMA_F32_16X16X4_F32 issue with an all-ones 16x4 A matrix:
// D[m][n] = Σ_k B[k][n] + C  ->  total = ΣD / 16 (each column replicated 16x).
// ---------------------------------------------------------------------------
__global__ __launch_bounds__(32) void final_kernel(const float* __restrict__ part_div,
                                                   const float* __restrict__ part_med,
                                                   float* __restrict__ out) {
    __shared__ float sm[32];
    int lane = threadIdx.x;
    float S[11];
#pragma unroll 1
    for (int k = 0; k < 11; ++k) {
        const float* p;
        int n;
        if (k < 4) { p = part_div + k * DIV_BLOCKS; n = DIV_BLOCKS; }
        else       { p = part_med + (k - 4) * MED_BLOCKS; n = MED_BLOCKS; }
#if defined(HAVE_WMMA_F32)
        v8f acc = {0.f, 0.f, 0.f, 0.f, 0.f, 0.f, 0.f, 0.f};
        v2f ones = {1.f, 1.f};
        for (int base = 0; base < n; base += 64) {
            v2f bv = *(const v2f*)(p + base + 2 * lane);   // 64 consecutive floats/wave
            acc = __builtin_amdgcn_wmma_f32_16x16x4_f32(
                /*neg_a=*/false, ones, /*neg_b=*/false, bv,
                /*c_mod=*/(short)0, acc, /*reuse_a=*/false, /*reuse_b=*/false);
        }
        float s = acc[0] + acc[1] + acc[2] + acc[3] + acc[4] + acc[5] + acc[6] + acc[7];
        sm[lane] = s;
        __syncthreads();
        float tot = 0.f;
        for (int i = 0; i < 32; ++i) tot += sm[i];
        S[k] = tot * (1.f / 16.f);
        __syncthreads();
#else
        float s = 0.f;
        for (int i = lane; i < n; i += 32) s += p[i];
        sm[lane] = s;
        __syncthreads();
        float tot = 0.f;
        for (int i = 0; i < 32; ++i) tot += sm[i];
        S[k] = tot;
        __syncthreads();
#endif
    }
    if (lane == 0) {
        const float ncell = (float)NCELL;
        float mp = S[0] / ncell, mt = S[2] / ncell;
        out[0] = mp;
        out[1] = S[1] / ncell - mp * mp;
        out[2] = mt;
        out[3] = S[3] / ncell - mt * mt;
        out[4] = (S[4] + S[5] + S[6]) / (float)NJ;
        out[5] = (S[7] + S[8] + S[9] + S[10]) / (float)V;
    }
}

// ---------------------------------------------------------------------------
// Host launcher
// ---------------------------------------------------------------------------
extern "C" void kernel_launch(void* const* d_in, const int* in_sizes, int n_in,
                              void* d_out, int out_size, void* d_ws, size_t ws_size,
                              hipStream_t stream) {
    (void)in_sizes; (void)n_in; (void)out_size; (void)ws_size;
    const float* pred_b  = (const float*)d_in[0];   // (1,3,96,96,96)
    const float* pred_z  = (const float*)d_in[1];   // (1,1,96,96,96)
    const float* targets = (const float*)d_in[2];   // (1,3,96,96,96)
    float* out = (float*)d_out;                     // 6 scalars

    // Workspace layout (floats): ~17.6 MB total
    float* ws   = (float*)d_ws;
    float* bxm  = ws;
    float* bym  = bxm + V;
    float* jx   = bym + V;
    float* jy   = jx + NJ;
    float* jz   = jy + NJ;
    float* pdiv = jz + NJ;                          // 4 * DIV_BLOCKS (padded)
    float* pmed = pdiv + 4 * DIV_BLOCKS;            // 7 * MED_BLOCKS

    prep_kernel<<<(V + 255) / 256, 256, 0, stream>>>(pred_b, targets, bxm, bym);
    j_kernel<<<(V + 255) / 256, 256, 0, stream>>>(bxm, bym, pred_b + 2 * V, jx, jy, jz);
    div_kernel<<<DIV_BLOCKS, 256, 0, stream>>>(pred_b, pred_z, targets, pdiv);

    dim3 mg(12, 12, 12);
    med_kernel<<<mg, 512, 0, stream>>>(jx,          NG, NC, NC, pmed + 0 * MED_BLOCKS);
    med_kernel<<<mg, 512, 0, stream>>>(jy,          NC, NG, NC, pmed + 1 * MED_BLOCKS);
    med_kernel<<<mg, 512, 0, stream>>>(jz,          NC, NC, NG, pmed + 2 * MED_BLOCKS);
    med_kernel<<<mg, 512, 0, stream>>>(bxm,         NG, NG, NG, pmed + 3 * MED_BLOCKS);
    med_kernel<<<mg, 512, 0, stream>>>(bym,         NG, NG, NG, pmed + 4 * MED_BLOCKS);
    med_kernel<<<mg, 512, 0, stream>>>(pred_b,      NG, NG, NG, pmed + 5 * MED_BLOCKS); // bx_p
    med_kernel<<<mg, 512, 0, stream>>>(pred_b + V,  NG, NG, NG, pmed + 6 * MED_BLOCKS); // by_p

    final_kernel<<<1, 32, 0, stream>>>(pdiv, pmed, out);
}